// GraphAttentionLayer_81595788689803
// MI455X (gfx1250) — compile-verified
//
#include <hip/hip_runtime.h>
#include <math.h>

#define N_NODES 1024
#define F_DIM   256
#define D_DIM   64
#define R_REL   11
#define E_EDGE  2048
#define ALPHA   0.2f

// Pair-interleaved LDS pitch for W[r]: 2*D_DIM data floats + 32 pad floats.
// P mod 64 == 32 -> half-wave (K+2 rows) b64 reads hit the other 32 banks.
#define B_PITCH 160

typedef float v2f __attribute__((ext_vector_type(2)));
typedef float v8f __attribute__((ext_vector_type(8)));

// ---------------- K0: p[r,f] = W[r,f,:].a[0:64], q[r,f] = W[r,f,:].a[64:128]
__global__ void pq_kernel(const float* __restrict__ W, const float* __restrict__ a,
                          float* __restrict__ p, float* __restrict__ q) {
    int r = blockIdx.x;
    int f = threadIdx.x;
    const float* wrow = W + ((size_t)r * F_DIM + f) * D_DIM;
    float sp = 0.f, sq = 0.f;
#pragma unroll 8
    for (int d = 0; d < D_DIM; ++d) {
        float w = wrow[d];
        sp += w * a[d];
        sq += w * a[D_DIM + d];
    }
    p[r * F_DIM + f] = sp;
    q[r * F_DIM + f] = sq;
}

// ---------------- K1: logit[e,f] = lrelu(x[src,f]*p[r,f] + x[dst,f]*q[r,f])
__global__ void logit_kernel(const float* __restrict__ x, const float* __restrict__ p,
                             const float* __restrict__ q, const int* __restrict__ src,
                             const int* __restrict__ dst, const int* __restrict__ rel,
                             float* __restrict__ logit) {
    int e = blockIdx.x, f = threadIdx.x;
    int s = src[e], d = dst[e], r = rel[e];
    float l = x[s * F_DIM + f] * p[r * F_DIM + f] + x[d * F_DIM + f] * q[r * F_DIM + f];
    l = (l >= 0.f) ? l : ALPHA * l;
    logit[(size_t)e * F_DIM + f] = l;
}

// ---------------- K2a: per-feature (column) max & exp-sum over E edges
__global__ void colstat_kernel(const float* __restrict__ logit,
                               float* __restrict__ colmax, float* __restrict__ colsum) {
    __shared__ float red[256];
    int f = blockIdx.x, t = threadIdx.x;
    float m = -3.402823466e38f;
#pragma unroll
    for (int i = 0; i < E_EDGE / 256; ++i)
        m = fmaxf(m, logit[(size_t)(t + i * 256) * F_DIM + f]);
    red[t] = m; __syncthreads();
    for (int s = 128; s > 0; s >>= 1) {
        if (t < s) red[t] = fmaxf(red[t], red[t + s]);
        __syncthreads();
    }
    m = red[0]; __syncthreads();
    float sum = 0.f;
#pragma unroll
    for (int i = 0; i < E_EDGE / 256; ++i)
        sum += __expf(logit[(size_t)(t + i * 256) * F_DIM + f] - m);
    red[t] = sum; __syncthreads();
    for (int s = 128; s > 0; s >>= 1) {
        if (t < s) red[t] += red[t + s];
        __syncthreads();
    }
    if (t == 0) { colmax[f] = m; colsum[f] = red[0]; }
}

// ---------------- K2b: in-place v[e,f] = softmax(logit)[e,f] * x[dst[e],f]
__global__ void attnv_kernel(float* __restrict__ logit, const float* __restrict__ colmax,
                             const float* __restrict__ colsum, const float* __restrict__ x,
                             const int* __restrict__ dst) {
    int e = blockIdx.x, f = threadIdx.x;
    float l = logit[(size_t)e * F_DIM + f];
    float attn = __expf(l - colmax[f]) / colsum[f];
    logit[(size_t)e * F_DIM + f] = attn * x[dst[e] * F_DIM + f];
}

// ---------------- K3: bin edges by relation (offsets + ordered edge list)
__global__ void bin_kernel(const int* __restrict__ rel, int* __restrict__ offs,
                           int* __restrict__ order) {
    __shared__ int scnt[R_REL];
    __shared__ int soff[R_REL + 1];
    int t = threadIdx.x;
    if (t < R_REL) scnt[t] = 0;
    __syncthreads();
    for (int e = t; e < E_EDGE; e += 256) atomicAdd(&scnt[rel[e]], 1);
    __syncthreads();
    if (t == 0) {
        int acc = 0;
        for (int r = 0; r < R_REL; ++r) { soff[r] = acc; acc += scnt[r]; }
        soff[R_REL] = acc;
    }
    __syncthreads();
    if (t < R_REL) scnt[t] = soff[t];   // running cursor per bin
    if (t <= R_REL) offs[t] = soff[t];
    __syncthreads();
    for (int e = t; e < E_EDGE; e += 256) {
        int pos = atomicAdd(&scnt[rel[e]], 1);
        order[pos] = e;
    }
}

// ---------------- K4: per-relation tiled WMMA GEMM: out[e,:] = v[e,:] @ W[r]
// grid = (16 tile-groups, R relations); block = 256 threads = 8 waves.
// Each wave computes one 16-edge x 64 (D) tile with V_WMMA_F32_16X16X4_F32.
// W[r] is staged in LDS pair-interleaved: ldsB[(f>>1)*B_PITCH + 2*d + (f&1)],
// so each B operand {W[f,col], W[f+1,col]} is one aligned ds_load_b64, and the
// pitch (160 mod 64 == 32 banks) makes the two half-waves bank-disjoint.
__global__ void gemm_kernel(const float* __restrict__ v, const float* __restrict__ W,
                            const int* __restrict__ offs, const int* __restrict__ order,
                            float* __restrict__ out) {
    extern __shared__ float ldsB[];  // (F_DIM/2) * B_PITCH floats = 80KB
    int r = blockIdx.y;

    // Stage + permute W[r]: read coalesced float4 (4 consecutive d of one f row),
    // scatter as 4 stride-2 DWORD stores.
    const float4* Wsrc = (const float4*)(W + (size_t)r * F_DIM * D_DIM);
    for (int i4 = threadIdx.x; i4 < (F_DIM * D_DIM) / 4; i4 += blockDim.x) {
        float4 w4 = Wsrc[i4];
        int f  = i4 >> 4;          // 16 float4 per f row
        int d0 = (i4 & 15) << 2;   // starting d
        float* dp = ldsB + (size_t)(f >> 1) * B_PITCH + (f & 1) + 2 * d0;
        dp[0] = w4.x; dp[2] = w4.y; dp[4] = w4.z; dp[6] = w4.w;
    }
    __syncthreads();

    int off = offs[r];
    int cnt = offs[r + 1] - off;
    int wave = threadIdx.x >> 5;
    int tile = blockIdx.x * (blockDim.x >> 5) + wave;   // up to 128 tiles/relation
    if (tile * 16 >= cnt) return;                        // wave-uniform exit

    int lane = threadIdx.x & 31;
    int m = lane & 15;        // A row / B column within 16
    int h = lane >> 4;        // half: K offset of 2
    int koff = h * 2;

    // A-row pointer (padded rows load a valid fallback row; their garbage C
    // rows are never stored — matmul rows are independent).
    int rowA = tile * 16 + m;
    int eidA = (rowA < cnt) ? order[off + rowA] : order[off];
    const float* arow = v + (size_t)eidA * F_DIM + koff;

    v8f c0 = {}, c1 = {}, c2 = {}, c3 = {};
    for (int fk = 0; fk < F_DIM; fk += 4) {
        v2f a = *(const v2f*)(arow + fk);               // K = fk+koff, fk+koff+1
        // pair row index: (fk+koff)/2 = fk/2 + h
        const float* bp = ldsB + (size_t)((fk >> 1) + h) * B_PITCH + 2 * m;
        v2f b0 = *(const v2f*)(bp + 2 * (0 * 16));
        v2f b1 = *(const v2f*)(bp + 2 * (1 * 16));
        v2f b2 = *(const v2f*)(bp + 2 * (2 * 16));
        v2f b3 = *(const v2f*)(bp + 2 * (3 * 16));
        c0 = __builtin_amdgcn_wmma_f32_16x16x4_f32(false, a, false, b0, (short)0, c0, false, false);
        c1 = __builtin_amdgcn_wmma_f32_16x16x4_f32(false, a, false, b1, (short)0, c1, false, false);
        c2 = __builtin_amdgcn_wmma_f32_16x16x4_f32(false, a, false, b2, (short)0, c2, false, false);
        c3 = __builtin_amdgcn_wmma_f32_16x16x4_f32(false, a, false, b3, (short)0, c3, false, false);
    }

    // C/D layout: VGPR vv, lane (m + 16h) holds element (M = vv + 8h, N = m + 16*nt)
    int eids[8];
#pragma unroll
    for (int vv = 0; vv < 8; ++vv) {
        int row = tile * 16 + vv + 8 * h;
        eids[vv] = (row < cnt) ? order[off + row] : -1;
    }
#pragma unroll
    for (int vv = 0; vv < 8; ++vv) {
        if (eids[vv] >= 0) {
            float* o = out + (size_t)eids[vv] * D_DIM;
            o[0 * 16 + m] = c0[vv];
            o[1 * 16 + m] = c1[vv];
            o[2 * 16 + m] = c2[vv];
            o[3 * 16 + m] = c3[vv];
        }
    }
}

extern "C" void kernel_launch(void* const* d_in, const int* in_sizes, int n_in,
                              void* d_out, int out_size, void* d_ws, size_t ws_size,
                              hipStream_t stream) {
    const float* x   = (const float*)d_in[0];
    const float* W   = (const float*)d_in[1];
    const float* a   = (const float*)d_in[2];
    const int*   src = (const int*)d_in[3];
    const int*   dst = (const int*)d_in[4];
    const int*   rel = (const int*)d_in[5];
    float* out = (float*)d_out;

    char* w = (char*)d_ws;
    auto carve = [&](size_t bytes) {
        char* pp = w;
        w += (bytes + 255) & ~(size_t)255;
        return pp;
    };
    float* p     = (float*)carve((size_t)R_REL * F_DIM * sizeof(float));
    float* q     = (float*)carve((size_t)R_REL * F_DIM * sizeof(float));
    float* cmax  = (float*)carve((size_t)F_DIM * sizeof(float));
    float* csum  = (float*)carve((size_t)F_DIM * sizeof(float));
    int*   offs  = (int*)carve((size_t)(R_REL + 1) * sizeof(int));
    int*   order = (int*)carve((size_t)E_EDGE * sizeof(int));
    float* logit = (float*)carve((size_t)E_EDGE * F_DIM * sizeof(float)); // reused as v

    pq_kernel     <<<dim3(R_REL),     256, 0, stream>>>(W, a, p, q);
    logit_kernel  <<<dim3(E_EDGE),    256, 0, stream>>>(x, p, q, src, dst, rel, logit);
    colstat_kernel<<<dim3(F_DIM),     256, 0, stream>>>(logit, cmax, csum);
    attnv_kernel  <<<dim3(E_EDGE),    256, 0, stream>>>(logit, cmax, csum, x, dst);
    bin_kernel    <<<dim3(1),         256, 0, stream>>>(rel, offs, order);
    gemm_kernel   <<<dim3(16, R_REL), 256,
                    (F_DIM / 2) * B_PITCH * sizeof(float), stream>>>
                  (logit, W, offs, order, out);
}